// OhemCrossEntropy2d_8375186227624
// MI455X (gfx1250) — compile-verified
//
#include <hip/hip_runtime.h>

// ---------------- problem constants (from reference) ----------------
#define NN   8
#define CC   19
#define HH   512
#define WW   1024
#define HSS  64            // H / FACTOR
#define WSS  128           // W / FACTOR
#define NS   (NN * HSS * WSS)   // 65536 downsampled samples
#define K_TH 3124               // MIN_KEPT/(8*8) - 1
#define THRESH 0.6f

// ---------------- CDNA5 TDM plumbing ----------------
typedef __attribute__((ext_vector_type(4))) unsigned v4u;
typedef __attribute__((ext_vector_type(8))) int      v8i;
typedef __attribute__((ext_vector_type(4))) int      v4i;

#if __has_include(<hip/amd_detail/amd_gfx1250_TDM.h>)
#define TDM_SIX_ARG 1
#endif

#define CPOL_RT 0   // regular temporal
#define CPOL_NT 1   // non-temporal (TH[2:0] = 1, scope = WGP)

static __device__ __forceinline__ unsigned rfl(unsigned x) {
  return (unsigned)__builtin_amdgcn_readfirstlane((int)x);
}

// LDS byte offset of a generic pointer that points into LDS
static __device__ __forceinline__ unsigned lds_off_of(const void* p) {
  return (unsigned)(unsigned long long)(__attribute__((address_space(3))) const char*)p;
}

// Issue a 2-D TENSOR_LOAD_TO_LDS: tile_d0 elements (contiguous f32) x tile_d1
// rows, row stride = stride0_elems (f32 units).  D# packed per CDNA5 ISA.
// CPOL is a template parameter so the builtin's cache-policy operand stays an ICE.
template <int CPOL>
static __device__ __forceinline__ void tdm_load_2d(unsigned long long gaddr,
                                                   unsigned lds_byte_off,
                                                   unsigned tile_d0,
                                                   unsigned tile_d1,
                                                   unsigned long long stride0_elems) {
  // group0: count=1 | lds_addr | global_addr[56:0] | type=2
  v4u g0;
  g0.x = rfl(1u);
  g0.y = rfl(lds_byte_off);
  g0.z = rfl((unsigned)(gaddr & 0xFFFFFFFFull));
  g0.w = rfl((unsigned)((gaddr >> 32) & 0x1FFFFFFull) | (2u << 30));
  // group1: data_size=4B; tensor dims == tile dims (no OOB path); dim0 stride
  const unsigned td0 = tile_d0, td1 = tile_d1;
  v8i g1;
  g1[0] = (int)rfl(2u << 16);                                   // data_size = 4B
  g1[1] = (int)rfl((td0 & 0xFFFFu) << 16);                      // tensor_dim0 lo16
  g1[2] = (int)rfl((td0 >> 16) | ((td1 & 0xFFFFu) << 16));      // d0 hi | d1 lo
  g1[3] = (int)rfl((td1 >> 16) | ((tile_d0 & 0xFFFFu) << 16));  // d1 hi | tile_dim0
  g1[4] = (int)rfl(tile_d1 & 0xFFFFu);                          // tile_dim1 (tile_dim2=0)
  g1[5] = (int)rfl((unsigned)(stride0_elems & 0xFFFFFFFFull));  // stride0 lo32
  g1[6] = (int)rfl((unsigned)((stride0_elems >> 32) & 0xFFFFull)); // stride0 hi16
  g1[7] = 0;
  v4i z4 = {0, 0, 0, 0};
#ifdef TDM_SIX_ARG
  v8i z8 = {0, 0, 0, 0, 0, 0, 0, 0};
  __builtin_amdgcn_tensor_load_to_lds(g0, g1, z4, z4, z8, CPOL);
#else
  __builtin_amdgcn_tensor_load_to_lds(g0, g1, z4, z4, CPOL);
#endif
}

// softmax probability of channel L at one LDS column; rowStride = f32 per channel row
static __device__ __forceinline__ float softmax_at(const float* col, int rowStride, int L) {
  float m = -3.402823466e38f;
#pragma unroll
  for (int c = 0; c < CC; ++c) m = fmaxf(m, col[c * rowStride]);
  float s = 0.0f;
#pragma unroll
  for (int c = 0; c < CC; ++c) s += __expf(col[c * rowStride] - m);
  return __expf(col[L * rowStride] - m) / s;
}

// ---------------- kernel 1: downsampled label-prob (pred_s) ----------------
// grid = NN*HSS blocks x 128 threads.  4 waves each TDM one (row, x-half) slab:
// LDS layout [slab(4)][c(19)][x(512)].  TH=RT so rows stay in the 192 MB L2
// for the full-res pass that re-reads them.
__global__ void ohem_downsample(const float* __restrict__ predict,
                                const int* __restrict__ target,
                                float* __restrict__ pred_s) {
  extern __shared__ float buf[];   // 4*19*512 f32 = 152 KB
  const int blk = blockIdx.x;
  const int n  = blk / HSS;
  const int ys = blk - n * HSS;

  const float yc = (float)ys * (511.0f / 63.0f);
  int y0 = (int)floorf(yc); if (y0 > HH - 1) y0 = HH - 1;
  const int y1 = (y0 + 1 < HH) ? (y0 + 1) : (HH - 1);
  const float wy = yc - (float)y0;

  const unsigned wave = (unsigned)threadIdx.x >> 5;   // 0..3
  const int r  = (int)(wave >> 1);                    // 0 -> row y0, 1 -> row y1
  const int hx = (int)(wave & 1u);                    // x half
  const int yr = r ? y1 : y0;

  const unsigned long long gaddr =
      (unsigned long long)(const void*)predict +
      4ull * (((unsigned long long)n * CC * HH + (unsigned long long)yr) * WW +
              (unsigned long long)(hx * 512));
  tdm_load_2d<CPOL_RT>(gaddr, lds_off_of(buf) + wave * (CC * 512u * 4u),
                       512u, (unsigned)CC, (unsigned long long)HH * WW);
  __builtin_amdgcn_s_wait_tensorcnt(0);
  __syncthreads();   // slabs are shared across waves

  const int xs = threadIdx.x;                         // 0..127
  const float xc = (float)xs * (1023.0f / 127.0f);
  int x0 = (int)floorf(xc); if (x0 > WW - 1) x0 = WW - 1;
  const int x1 = (x0 + 1 < WW) ? (x0 + 1) : (WW - 1);
  const float wx = xc - (float)x0;

  int yi = (int)rintf(yc); if (yi > HH - 1) yi = HH - 1;
  int xi = (int)rintf(xc); if (xi > WW - 1) xi = WW - 1;
  const int L = target[(n * HH + yi) * WW + xi];

  // corner (rr, x) lives in slab rr*2 + (x>>9) at column x&511
  const float* c00 = buf + (((0 * 2 + (x0 >> 9)) * CC) * 512) + (x0 & 511);
  const float* c01 = buf + (((0 * 2 + (x1 >> 9)) * CC) * 512) + (x1 & 511);
  const float* c10 = buf + (((1 * 2 + (x0 >> 9)) * CC) * 512) + (x0 & 511);
  const float* c11 = buf + (((1 * 2 + (x1 >> 9)) * CC) * 512) + (x1 & 511);
  const float p00 = softmax_at(c00, 512, L);
  const float p01 = softmax_at(c01, 512, L);
  const float p10 = softmax_at(c10, 512, L);
  const float p11 = softmax_at(c11, 512, L);
  const float top = p00 * (1.0f - wx) + p01 * wx;
  const float bot = p10 * (1.0f - wx) + p11 * wx;
  pred_s[(n * HSS + ys) * WSS + xs] = top * (1.0f - wy) + bot * wy;
}

// ---------------- kernel 2: rank-K_TH via 4-round radix select ----------------
// pred_s are positive floats -> uint bit pattern is order-preserving.
__global__ void ohem_select(const float* __restrict__ pred_s,
                            float* __restrict__ thr_out) {
  __shared__ unsigned hist[256];
  __shared__ unsigned sh_prefix, sh_krem;
  const unsigned t = threadIdx.x;   // 256 threads
  if (t == 0) { sh_prefix = 0u; sh_krem = K_TH; }
  __syncthreads();
  unsigned mask = 0u;
  for (int shift = 24; shift >= 0; shift -= 8) {
    hist[t] = 0u;
    __syncthreads();
    const unsigned prefix = sh_prefix;
    for (int i = 0; i < NS / 256; ++i) {
      if (i + 8 < NS / 256) __builtin_prefetch(&pred_s[t + 256u * (i + 8)], 0, 0);
      const unsigned bits = __float_as_uint(pred_s[t + 256u * i]);
      if ((bits & mask) == prefix) atomicAdd(&hist[(bits >> shift) & 255u], 1u);
    }
    __syncthreads();
    if (t == 0) {
      unsigned k = sh_krem, cum = 0u, b = 0u;
      for (; b < 256u; ++b) {
        const unsigned h = hist[b];
        if (cum + h > k) break;
        cum += h;
      }
      sh_prefix = prefix | (b << shift);
      sh_krem   = k - cum;
    }
    __syncthreads();
    mask |= (0xFFu << shift);
  }
  if (t == 0) {
    const float kth = __uint_as_float(sh_prefix);
    thr_out[0] = (kth > THRESH) ? kth : THRESH;
  }
}

// ---------------- kernel 3: full-res keep/ignore pass ----------------
// grid = NN*HH blocks x 256 threads; one image row per block.  Each of 8 waves
// owns a private 128-px chunk, split into two 19x64 TDM tiles that are issued
// back-to-back and consumed pipelined:
//   issue T0, issue T1, s_wait_tensorcnt<=1, compute T0, s_wait_tensorcnt 0,
//   compute T1  (TENSORcnt ops complete in order within a wave).
// TH=NT: predict is streamed exactly once here, keep it out of L2.
__global__ void ohem_mark(const float* __restrict__ predict,
                          const int* __restrict__ target,
                          const float* __restrict__ thr_p,
                          float* __restrict__ out) {
  extern __shared__ float smem[];   // [wave(8)][tile(2)][c(19)][64] = 76 KB
  const int blk = blockIdx.x;
  const int n = blk / HH;
  const int y = blk - n * HH;
  const unsigned wave = (unsigned)threadIdx.x >> 5;   // 0..7
  const unsigned lane = (unsigned)threadIdx.x & 31u;
  const int rowBase = (n * HH + y) * WW;

  const unsigned long long rowAddr =
      (unsigned long long)(const void*)predict +
      4ull * (((unsigned long long)n * CC * HH + (unsigned long long)y) * WW);
  const unsigned ldsBase = lds_off_of(smem) + wave * (2u * CC * 64u * 4u);

  tdm_load_2d<CPOL_NT>(rowAddr + 4ull * (wave * 128u),
                       ldsBase, 64u, (unsigned)CC, (unsigned long long)HH * WW);
  tdm_load_2d<CPOL_NT>(rowAddr + 4ull * (wave * 128u + 64u),
                       ldsBase + CC * 64u * 4u, 64u, (unsigned)CC,
                       (unsigned long long)HH * WW);

  const float thr = thr_p[0];
  const float* wbuf = smem + wave * (2u * CC * 64u);

  // ---- tile 0: its DMA is done once TENSORcnt <= 1 ----
  __builtin_amdgcn_s_wait_tensorcnt(1);
#pragma unroll
  for (int j = 0; j < 2; ++j) {
    const int xi = (int)lane + 32 * j;                // 0..63 in tile
    const int x  = (int)wave * 128 + xi;
    const int L  = target[rowBase + x];
    const float p = softmax_at(wbuf + xi, 64, L);
    out[rowBase + x] = ((p <= thr) && (L >= 0)) ? (float)L : -1.0f;
  }

  // ---- tile 1 ----
  __builtin_amdgcn_s_wait_tensorcnt(0);
  const float* tbuf = wbuf + CC * 64u;
#pragma unroll
  for (int j = 0; j < 2; ++j) {
    const int xi = (int)lane + 32 * j;
    const int x  = (int)wave * 128 + 64 + xi;
    const int L  = target[rowBase + x];
    const float p = softmax_at(tbuf + xi, 64, L);
    out[rowBase + x] = ((p <= thr) && (L >= 0)) ? (float)L : -1.0f;
  }
}

// ---------------- host-side launch ----------------
extern "C" void kernel_launch(void* const* d_in, const int* in_sizes, int n_in,
                              void* d_out, int out_size, void* d_ws, size_t ws_size,
                              hipStream_t stream) {
  (void)in_sizes; (void)n_in; (void)out_size; (void)ws_size;
  const float* predict = (const float*)d_in[0];
  const int*   target  = (const int*)d_in[1];
  float* wsf    = (float*)d_ws;
  float* pred_s = wsf;             // NS floats (256 KB)
  float* thr    = wsf + NS;        // 1 float
  float* out    = (float*)d_out;   // (8,512,1024) labels as float (-1 = ignore)

  ohem_downsample<<<dim3(NN * HSS), dim3(128), 4 * CC * 512 * 4, stream>>>(predict, target, pred_s);
  ohem_select<<<dim3(1), dim3(256), 0, stream>>>(pred_s, thr);
  ohem_mark<<<dim3(NN * HH), dim3(256), 8 * 2 * CC * 64 * 4, stream>>>(predict, target, thr, out);
}